// ZooBP_43293270343692
// MI455X (gfx1250) — compile-verified
//
#include <hip/hip_runtime.h>
#include <hip/hip_bf16.h>

// ---- problem constants (match reference) ----
#define DD     18
#define EPSV   0.1f
#define CCV    0.01f
#define DIMA   4
#define DIMP   18
#define SCALE  (EPSV / 18.0f)   // eps / D

typedef __attribute__((ext_vector_type(2))) float v2f;
typedef __attribute__((ext_vector_type(8))) float v8f;

__device__ __forceinline__ void atomic_add_f32(float* p, float v) {
    // relaxed, device scope -> native global_atomic_add_f32
    __hip_atomic_fetch_add(p, v, __ATOMIC_RELAXED, __HIP_MEMORY_SCOPE_AGENT);
}

// ---------------------------------------------------------------------------
// Init: Ya[NA,4] = Xa*cm_a ; Yp[NP,20 padded] = Xp ;
//       Z0[NP,4] = Yp0 @ H_pa[:,0:4] (closed form from mask/label)
// ---------------------------------------------------------------------------
__global__ void zoobp_init(const int* __restrict__ maska, const int* __restrict__ labela,
                           const int* __restrict__ maskp, const int* __restrict__ labelp,
                           const float* __restrict__ Hpa,
                           float* __restrict__ Ya, float* __restrict__ Yp,
                           float* __restrict__ Z, int NA, int NP) {
    int i = blockIdx.x * blockDim.x + threadIdx.x;
    if (i < NA) {
        float ma = (maska[i] != 0) ? 1.f : 0.f;
        int   la = labela[i];
        #pragma unroll
        for (int c = 0; c < 4; ++c)
            Ya[(size_t)i * 4 + c] = ma * ((c == la) ? (DIMA * CCV - CCV) : -CCV);
    } else if (i < NA + NP) {
        int p = i - NA;
        float mp = (maskp[p] != 0) ? 1.f : 0.f;
        int   lp = labelp[p];
        float z0 = 0.f, z1 = 0.f, z2 = 0.f, z3 = 0.f;
        #pragma unroll
        for (int k = 0; k < DD; ++k) {
            float yk = mp * ((k == lp) ? (DIMP * CCV - CCV) : -CCV);
            Yp[(size_t)p * 20 + k] = yk;
            z0 += yk * Hpa[k * DD + 0];
            z1 += yk * Hpa[k * DD + 1];
            z2 += yk * Hpa[k * DD + 2];
            z3 += yk * Hpa[k * DD + 3];
        }
        Yp[(size_t)p * 20 + 18] = 0.f;
        Yp[(size_t)p * 20 + 19] = 0.f;
        float* zp = Z + (size_t)p * 4;
        zp[0] = z0; zp[1] = z1; zp[2] = z2; zp[3] = z3;
    }
}

// ---------------------------------------------------------------------------
// Edge pass (symmetric, 4 cols each way):
//   Sp[dst] += w*Ya[src] ; Sa[src] += w*Z[dst]
// ---------------------------------------------------------------------------
__global__ void zoobp_edge(const int* __restrict__ esrc, const int* __restrict__ edst,
                           const float* __restrict__ w,
                           const float* __restrict__ Ya, const float* __restrict__ Z,
                           float* __restrict__ Sa, float* __restrict__ Sp, int E) {
    int e = blockIdx.x * blockDim.x + threadIdx.x;
    if (e >= E) return;
    // stream the edge arrays ahead (lowers to global_prefetch_b8)
    __builtin_prefetch(esrc + e + 4096, 0, 1);
    __builtin_prefetch(edst + e + 4096, 0, 1);
    __builtin_prefetch(w    + e + 4096, 0, 1);

    int   s  = esrc[e];
    int   d  = edst[e];
    float we = w[e];

    const float4 ya = *reinterpret_cast<const float4*>(Ya + (size_t)s * 4);
    const float4 zd = *reinterpret_cast<const float4*>(Z  + (size_t)d * 4);

    float* spd = Sp + (size_t)d * 4;
    atomic_add_f32(spd + 0, we * ya.x);
    atomic_add_f32(spd + 1, we * ya.y);
    atomic_add_f32(spd + 2, we * ya.z);
    atomic_add_f32(spd + 3, we * ya.w);

    float* sas = Sa + (size_t)s * 4;
    atomic_add_f32(sas + 0, we * zd.x);
    atomic_add_f32(sas + 1, we * zd.y);
    atomic_add_f32(sas + 2, we * zd.z);
    atomic_add_f32(sas + 3, we * zd.w);
}

// ---------------------------------------------------------------------------
// Paper update (fused): one wave32 per 16 papers.
//   1) Yp = Xp + SCALE*(Sp[:,0:4] @ H_ap[0:4,0:18])   -> 2x V_WMMA_F32_16X16X4_F32
//   2) transpose 16x18 tile through LDS (D-layout -> A-layout)
//   3) Z = Yp @ H_pa[:,0:4]                           -> 5x chained WMMA (K=18)
// ---------------------------------------------------------------------------
__global__ void zoobp_paper(const float* __restrict__ Sp,
                            const float* __restrict__ Hap, const float* __restrict__ Hpa,
                            const int* __restrict__ maskp, const int* __restrict__ labelp,
                            float* __restrict__ Yp, float* __restrict__ Z, int NP) {
    __shared__ float tile[8][16][20];            // 8 waves/block, stride-20 rows: bank-conflict-free
    int wib   = threadIdx.x >> 5;
    int gwave = (blockIdx.x * blockDim.x + threadIdx.x) >> 5;
    int lane  = threadIdx.x & 31;
    int row0  = gwave * 16;
    if (row0 >= NP) return;                      // wave-uniform exit

    int m    = lane & 15;
    int half = lane >> 4;
    int r    = row0 + m;
    int ka   = half * 2;                         // A layout: v0 = K{0|2}, v1 = K{1|3}
    int kb   = ka + 1;

    v2f a;
    a.x = (r < NP) ? Sp[(size_t)r * 4 + ka] : 0.f;
    a.y = (r < NP) ? Sp[(size_t)r * 4 + kb] : 0.f;

    v2f b0, b1;
    b0.x = Hap[ka * DD + m];
    b0.y = Hap[kb * DD + m];
    int n1 = 16 + m;
    b1.x = (n1 < DD) ? Hap[ka * DD + n1] : 0.f;
    b1.y = (n1 < DD) ? Hap[kb * DD + n1] : 0.f;

    v8f c0 = {}; v8f c1 = {};
    c0 = __builtin_amdgcn_wmma_f32_16x16x4_f32(false, a, false, b0, (short)0, c0, false, false);
    c1 = __builtin_amdgcn_wmma_f32_16x16x4_f32(false, a, false, b1, (short)0, c1, false, false);

    // finalize Yp, write to global and to the LDS transpose tile
    #pragma unroll
    for (int j = 0; j < 8; ++j) {
        int rj = j + half * 8;                   // D layout: VGPR j -> row j (+8 for hi lanes)
        int rr = row0 + rj;
        if (rr < NP) {
            float mp = (maskp[rr] != 0) ? 1.f : 0.f;
            int   lp = labelp[rr];
            float y0 = mp * ((m == lp) ? (DIMP * CCV - CCV) : -CCV) + SCALE * c0[j];
            Yp[(size_t)rr * 20 + m] = y0;
            tile[wib][rj][m] = y0;
            if (n1 < DD) {
                float y1 = mp * ((n1 == lp) ? (DIMP * CCV - CCV) : -CCV) + SCALE * c1[j];
                Yp[(size_t)rr * 20 + n1] = y1;
                tile[wib][rj][n1] = y1;
            }
        }
    }
    // intra-wave LDS write->read is ordered; compiler inserts s_wait_dscnt for the reads

    // Z = Yp_tile @ H_pa[:,0:4] : K=18 chain of 5 WMMAs (A from LDS in A-layout)
    v8f acc = {};
    #pragma unroll
    for (int kt = 0; kt < 5; ++kt) {
        int kx = kt * 4 + half * 2;
        int ky = kx + 1;
        v2f a2, b2;
        a2.x = (kx < DD) ? tile[wib][m][kx] : 0.f;
        a2.y = (ky < DD) ? tile[wib][m][ky] : 0.f;
        b2.x = (kx < DD) ? Hpa[kx * DD + m] : 0.f;
        b2.y = (ky < DD) ? Hpa[ky * DD + m] : 0.f;
        acc = __builtin_amdgcn_wmma_f32_16x16x4_f32(false, a2, false, b2, (short)0, acc, false, false);
    }

    if (m < 4) {                                 // only first DIMA columns of Z are live
        #pragma unroll
        for (int j = 0; j < 8; ++j) {
            int rr = row0 + j + half * 8;
            if (rr < NP) Z[(size_t)rr * 4 + m] = acc[j];
        }
    }
}

// ---------------------------------------------------------------------------
// Author update (pure elementwise now): Ya = Xa + SCALE * Sa
// ---------------------------------------------------------------------------
__global__ void zoobp_author(const float* __restrict__ Sa,
                             const int* __restrict__ maska, const int* __restrict__ labela,
                             float* __restrict__ Ya, int NA) {
    int a = blockIdx.x * blockDim.x + threadIdx.x;
    if (a >= NA) return;
    float ma = (maska[a] != 0) ? 1.f : 0.f;
    int   la = labela[a];
    const float4 s4 = *reinterpret_cast<const float4*>(Sa + (size_t)a * 4);
    float4 y;
    y.x = ma * ((0 == la) ? (DIMA * CCV - CCV) : -CCV) + SCALE * s4.x;
    y.y = ma * ((1 == la) ? (DIMA * CCV - CCV) : -CCV) + SCALE * s4.y;
    y.z = ma * ((2 == la) ? (DIMA * CCV - CCV) : -CCV) + SCALE * s4.z;
    y.w = ma * ((3 == la) ? (DIMA * CCV - CCV) : -CCV) + SCALE * s4.w;
    *reinterpret_cast<float4*>(Ya + (size_t)a * 4) = y;
}

// ---------------------------------------------------------------------------
// Expand internal buffers into d_out: [NA*18 | NP*18] flat, Ya cols>=4 are 0
// ---------------------------------------------------------------------------
__global__ void zoobp_out(const float* __restrict__ Ya, const float* __restrict__ Yp,
                          float* __restrict__ out, int NA, int NP) {
    int i = blockIdx.x * blockDim.x + threadIdx.x;
    int totA = NA * DD;
    if (i < totA) {
        int a = i / DD, c = i - a * DD;
        out[i] = (c < 4) ? Ya[(size_t)a * 4 + c] : 0.f;
    } else if (i < totA + NP * DD) {
        int j = i - totA;
        int p = j / DD, c = j - p * DD;
        out[i] = Yp[(size_t)p * 20 + c];
    }
}

extern "C" void kernel_launch(void* const* d_in, const int* in_sizes, int n_in,
                              void* d_out, int out_size, void* d_ws, size_t ws_size,
                              hipStream_t stream) {
    const float* Hap    = (const float*)d_in[0];
    const float* Hpa    = (const float*)d_in[1];
    const float* w      = (const float*)d_in[2];
    const int*   esrc   = (const int*)d_in[3];
    const int*   edst   = (const int*)d_in[4];
    const int*   maska  = (const int*)d_in[5];
    const int*   labela = (const int*)d_in[6];
    const int*   maskp  = (const int*)d_in[7];
    const int*   labelp = (const int*)d_in[8];

    const int E  = in_sizes[2];
    const int NA = in_sizes[5];
    const int NP = in_sizes[7];
    float* out = (float*)d_out;

    // workspace carve (all offsets multiples of 16 bytes)
    float* Ya = (float*)d_ws;                  // NA*4
    float* Z  = Ya + (size_t)NA * 4;           // NP*4
    float* Yp = Z  + (size_t)NP * 4;           // NP*20 (stride-20 -> 16B-aligned rows)
    float* Sa = Yp + (size_t)NP * 20;          // NA*4
    float* Sp = Sa + (size_t)NA * 4;           // NP*4
    // total = 8*NA + 28*NP floats ≈ 59 MB

    const int B = 256;
    int tInit = NA + NP;
    zoobp_init<<<(tInit + B - 1) / B, B, 0, stream>>>(maska, labela, maskp, labelp, Hpa,
                                                      Ya, Yp, Z, NA, NP);

    const int pthreads = ((NP + 15) / 16) * 32;   // one wave32 per 16 papers

    for (int it = 0; it < 5; ++it) {
        hipMemsetAsync(Sa, 0, (size_t)NA * 4 * sizeof(float), stream);
        hipMemsetAsync(Sp, 0, (size_t)NP * 4 * sizeof(float), stream);
        zoobp_edge<<<(E + B - 1) / B, B, 0, stream>>>(esrc, edst, w, Ya, Z, Sa, Sp, E);
        zoobp_paper<<<(pthreads + B - 1) / B, B, 0, stream>>>(Sp, Hap, Hpa, maskp, labelp,
                                                              Yp, Z, NP);
        zoobp_author<<<(NA + B - 1) / B, B, 0, stream>>>(Sa, maska, labela, Ya, NA);
    }

    int tOut = (NA + NP) * DD;
    zoobp_out<<<(tOut + B - 1) / B, B, 0, stream>>>(Ya, Yp, out, NA, NP);
}